// HashEncoding_10273561772148
// MI455X (gfx1250) — compile-verified
//
#include <hip/hip_runtime.h>
#include <cmath>

// ---------------------------------------------------------------------------
// Instant-NGP style multiresolution hash encoding, tuned for MI455X (gfx1250).
//
// Key design points (reasoned from MI455X specs, compile-only):
//  * 16 hash tables total 64 MB -> fully L2-resident (192 MB global L2).
//    Table gathers use default (RT) cache policy so they stay hot in L2/WGP$.
//  * Streaming traffic (x: 24 MB read, out: 256 MB write) uses non-temporal
//    loads/stores so it does not evict the tables from L2.
//  * One thread per point, all 16 levels fully unrolled -> up to 128
//    independent global_load_b64 gathers in flight per thread.
//  * Output rows are 128 B/point (uncoalescible per-thread), so each block
//    stages its 256x32-float tile in LDS (stride padded to 36 floats) and
//    performs a fully-coalesced cooperative float4 non-temporal store pass.
//  * No WMMA: the per-point contraction is 1x8 . 8x2 with both operands
//    unique per point -- no shared matrix operand exists, so matrix ops
//    cannot help; the kernel is L2-gather bound (~2 GB gathers, ~350 MB HBM).
// ---------------------------------------------------------------------------

#define N_LEVELS   16
#define LOG2_HASH  19
#define HASH_SIZE  (1u << LOG2_HASH)
#define HASH_MASK  (HASH_SIZE - 1u)
#define PRIME1     2654435761u
#define PRIME2     805459861u

#define TPB        256
#define LDS_PAD    36   // floats per point row in LDS: 32 used + 4 pad (bank spread, 16B align)

typedef float v4f __attribute__((ext_vector_type(4)));   // native vector: OK for nontemporal builtins

struct ResArr { int r[N_LEVELS]; };

__global__ __launch_bounds__(TPB)
void HashEncoding_10273561772148_kernel(const float* __restrict__ x,
                                        const float* __restrict__ tables,
                                        float* __restrict__ out,
                                        int N, ResArr rs)
{
    __shared__ float tile[TPB * LDS_PAD];

    const int tid = threadIdx.x;
    const int p   = blockIdx.x * TPB + tid;

    if (p < N) {
        // Streamed, non-temporal read of the 3 coords (keep L2 for tables).
        const float* xp = x + (size_t)3 * p;
        float px = __builtin_nontemporal_load(xp + 0);
        float py = __builtin_nontemporal_load(xp + 1);
        float pz = __builtin_nontemporal_load(xp + 2);

        const float hi = (float)(1.0 - 1e-6);   // matches jnp.clip upper bound
        px = fminf(fmaxf(px, 0.0f), hi);
        py = fminf(fmaxf(py, 0.0f), hi);
        pz = fminf(fmaxf(pz, 0.0f), hi);

        #pragma unroll
        for (int L = 0; L < N_LEVELS; ++L) {
            const float res = (float)rs.r[L];
            const float sx = px * res, sy = py * res, sz = pz * res;
            const float fx = floorf(sx), fy = floorf(sy), fz = floorf(sz);
            const float wx = sx - fx,   wy = sy - fy,   wz = sz - fz;

            const unsigned ix = (unsigned)(int)fx;
            const unsigned iy = (unsigned)(int)fy;
            const unsigned iz = (unsigned)(int)fz;

            // (c+1)*prime == c*prime + prime  (mult does not distribute over
            // XOR, but per-axis terms do): 2 mults + 3 adds for all 6 terms.
            const unsigned ax0 = ix;                // prime0 == 1
            const unsigned ax1 = ix + 1u;
            const unsigned by0 = iy * PRIME1;
            const unsigned by1 = by0 + PRIME1;
            const unsigned cz0 = iz * PRIME2;
            const unsigned cz1 = cz0 + PRIME2;

            const unsigned h000 = (ax0 ^ by0 ^ cz0) & HASH_MASK;
            const unsigned h001 = (ax0 ^ by0 ^ cz1) & HASH_MASK;
            const unsigned h010 = (ax0 ^ by1 ^ cz0) & HASH_MASK;
            const unsigned h011 = (ax0 ^ by1 ^ cz1) & HASH_MASK;
            const unsigned h100 = (ax1 ^ by0 ^ cz0) & HASH_MASK;
            const unsigned h101 = (ax1 ^ by0 ^ cz1) & HASH_MASK;
            const unsigned h110 = (ax1 ^ by1 ^ cz0) & HASH_MASK;
            const unsigned h111 = (ax1 ^ by1 ^ cz1) & HASH_MASK;

            // 8 independent b64 gathers, L2-resident table (default RT hint).
            const float2* __restrict__ tbl =
                (const float2*)tables + (size_t)L * HASH_SIZE;
            const float2 f000 = tbl[h000];
            const float2 f001 = tbl[h001];
            const float2 f010 = tbl[h010];
            const float2 f011 = tbl[h011];
            const float2 f100 = tbl[h100];
            const float2 f101 = tbl[h101];
            const float2 f110 = tbl[h110];
            const float2 f111 = tbl[h111];

            const float wx0 = 1.0f - wx, wy0 = 1.0f - wy, wz0 = 1.0f - wz;
            const float w00 = wx0 * wy0, w01 = wx0 * wy;
            const float w10 = wx  * wy0, w11 = wx  * wy;
            const float w000 = w00 * wz0, w001 = w00 * wz;
            const float w010 = w01 * wz0, w011 = w01 * wz;
            const float w100 = w10 * wz0, w101 = w10 * wz;
            const float w110 = w11 * wz0, w111 = w11 * wz;

            float o0 = w000 * f000.x;
            float o1 = w000 * f000.y;
            o0 = fmaf(w001, f001.x, o0);  o1 = fmaf(w001, f001.y, o1);
            o0 = fmaf(w010, f010.x, o0);  o1 = fmaf(w010, f010.y, o1);
            o0 = fmaf(w011, f011.x, o0);  o1 = fmaf(w011, f011.y, o1);
            o0 = fmaf(w100, f100.x, o0);  o1 = fmaf(w100, f100.y, o1);
            o0 = fmaf(w101, f101.x, o0);  o1 = fmaf(w101, f101.y, o1);
            o0 = fmaf(w110, f110.x, o0);  o1 = fmaf(w110, f110.y, o1);
            o0 = fmaf(w111, f111.x, o0);  o1 = fmaf(w111, f111.y, o1);

            // Stage this level's pair in LDS (8B-aligned -> ds_store_b64).
            tile[tid * LDS_PAD + 2 * L + 0] = o0;
            tile[tid * LDS_PAD + 2 * L + 1] = o1;
        }
    } else {
        #pragma unroll
        for (int L = 0; L < N_LEVELS; ++L) {
            tile[tid * LDS_PAD + 2 * L + 0] = 0.0f;
            tile[tid * LDS_PAD + 2 * L + 1] = 0.0f;
        }
    }

    __syncthreads();

    // Cooperative, fully-coalesced non-temporal store of the 256x32 tile.
    // Tile = 8192 floats = 2048 float4; 8 float4 per thread.
    const size_t blockPt = (size_t)blockIdx.x * TPB;
    float* outBase = out + blockPt * (2 * N_LEVELS);
    #pragma unroll
    for (int i = 0; i < 8; ++i) {
        const int q   = i * TPB + tid;      // float4 slot within tile
        const int pt  = q >> 3;             // point within block
        const int col = (q & 7) << 2;       // float column (0,4,...,28)
        if (blockPt + (size_t)pt < (size_t)N) {
            const v4f v = *(const v4f*)&tile[pt * LDS_PAD + col];
            __builtin_nontemporal_store(v, (v4f*)outBase + q);
        }
    }
}

extern "C" void kernel_launch(void* const* d_in, const int* in_sizes, int n_in,
                              void* d_out, int out_size, void* d_ws, size_t ws_size,
                              hipStream_t stream) {
    const float* x      = (const float*)d_in[0];
    const float* tables = (const float*)d_in[1];
    float*       out    = (float*)d_out;

    const int N = in_sizes[0] / 3;

    // Same double-precision resolution schedule as the reference.
    ResArr rs;
    const double b = std::exp((std::log(512.0) - std::log(16.0)) /
                              (double)(N_LEVELS - 1));
    for (int i = 0; i < N_LEVELS; ++i)
        rs.r[i] = (int)std::ceil(16.0 * std::pow(b, (double)i));

    const int grid = (N + TPB - 1) / TPB;
    HashEncoding_10273561772148_kernel<<<grid, TPB, 0, stream>>>(
        x, tables, out, N, rs);
}